// KeyTokenSelector_19516331393661
// MI455X (gfx1250) — compile-verified
//
#include <hip/hip_runtime.h>
#include <hip/hip_bf16.h>
#include <math.h>

typedef __attribute__((ext_vector_type(2))) float v2f;
typedef __attribute__((ext_vector_type(8))) float v8f;

#define B_    128
#define N_    197
#define DV_   1024
#define H_    16
#define L_    77
#define DT_   768
#define TOPK_IMG 39      /* int(197*0.2) */
#define MAXSEL_IMG 64    /* padded selection slots (multiple of 4) */
#define MAXSEL_TXT 80
#define CLS_ID_ 49406
#define EOS_ID_ 49407

// ---------------------------------------------------------------------------
// K1: patch importance = head-averaged CLS attention row, CLS self-attn zeroed.
// Reads only img_attn[b,h,0,:] (1.6 MB of the 318 MB tensor).
// ---------------------------------------------------------------------------
__global__ void k_patch_imp(const float* __restrict__ attn, float* __restrict__ pimp) {
  int b = blockIdx.x;
  int n = threadIdx.x;
  if (n >= N_) return;
  const float* base = attn + (size_t)b * H_ * N_ * N_ + n;  // row 0 of each head
  float s = 0.f;
#pragma unroll
  for (int h = 0; h < H_; ++h) s += base[(size_t)h * N_ * N_];
  s *= (1.0f / (float)H_);
  if (n == 0) s = 0.f;
  pimp[b * N_ + n] = s;
}

// ---------------------------------------------------------------------------
// K2: text token L2 norms (b128 loads), zeroed for invalid (id==0) tokens.
// 192 threads * float4 == 768 elements per row.
// ---------------------------------------------------------------------------
__global__ void k_text_norm(const float* __restrict__ tt, const int* __restrict__ ids,
                            float* __restrict__ timp) {
  int l = blockIdx.x, b = blockIdx.y;
  const float4* row = (const float4*)(tt + ((size_t)b * L_ + l) * DT_);
  int t = threadIdx.x;  // 0..191
  float4 v = row[t];
  float s = v.x * v.x + v.y * v.y + v.z * v.z + v.w * v.w;
  __shared__ float red[192];
  red[t] = s;
  __syncthreads();
  if (t < 64) red[t] += red[t + 64] + red[t + 128];
  __syncthreads();
  for (int off = 32; off > 0; off >>= 1) {
    if (t < off) red[t] += red[t + off];
    __syncthreads();
  }
  if (t == 0) {
    float nv = sqrtf(red[0]);
    if (ids[b * L_ + l] == 0) nv = 0.f;
    timp[b * L_ + l] = nv;
  }
}

// ---------------------------------------------------------------------------
// K3a: image top-k mask via stable descending rank; emit mask floats +
// compacted selection list (padded with -1) + 1/count.
// ---------------------------------------------------------------------------
__global__ void k_img_select(const float* __restrict__ pimp, float* __restrict__ mask_out,
                             int* __restrict__ sel, int* __restrict__ cnt_out,
                             float* __restrict__ invc) {
  int b = blockIdx.x;
  __shared__ float x[N_];
  __shared__ int cnt;
  __shared__ int slist[MAXSEL_IMG];
  int t = threadIdx.x;
  if (t == 0) cnt = 0;
  if (t < N_) x[t] = pimp[b * N_ + t];
  __syncthreads();
  if (t < N_) {
    float xi = x[t];
    int rank = 0;
    for (int j = 0; j < N_; ++j) {
      float xj = x[j];
      rank += (int)((xj > xi) || (xj == xi && j < t));   // stable argsort rank
    }
    bool m = (rank < TOPK_IMG) || (t == 0);              // always keep CLS
    mask_out[b * N_ + t] = m ? 1.0f : 0.0f;
    if (m) { int p = atomicAdd(&cnt, 1); slist[p] = t; }
  }
  __syncthreads();
  int c = cnt;
  for (int i = c + t; i < MAXSEL_IMG; i += blockDim.x) slist[i] = -1;
  __syncthreads();
  for (int i = t; i < MAXSEL_IMG; i += blockDim.x) sel[b * MAXSEL_IMG + i] = slist[i];
  if (t == 0) { cnt_out[b] = c; invc[b] = 1.0f / (float)c; }
}

// ---------------------------------------------------------------------------
// K3b: text variable-k top-k mask (+CLS/EOS) and selection list.
// ---------------------------------------------------------------------------
__global__ void k_txt_select(const float* __restrict__ timp, const int* __restrict__ ids,
                             float* __restrict__ mask_out, int* __restrict__ sel,
                             int* __restrict__ cnt_out, float* __restrict__ invc) {
  int b = blockIdx.x;
  __shared__ float x[L_];
  __shared__ int idsh[L_];
  __shared__ int vcnt;
  __shared__ int cnt;
  __shared__ int slist[MAXSEL_TXT];
  int t = threadIdx.x;
  if (t == 0) { cnt = 0; vcnt = 0; }
  __syncthreads();
  if (t < L_) {
    x[t] = timp[b * L_ + t];
    int id = ids[b * L_ + t];
    idsh[t] = id;
    if (id != 0) atomicAdd(&vcnt, 1);
  }
  __syncthreads();
  int vc = vcnt; if (vc < 1) vc = 1;
  int k = (int)((float)vc * 0.2f);
  if (k < 1) k = 1;
  if (k > vc) k = vc;
  if (t < L_) {
    float xi = x[t];
    int rank = 0;
    for (int j = 0; j < L_; ++j) {
      float xj = x[j];
      rank += (int)((xj > xi) || (xj == xi && j < t));
    }
    int id = idsh[t];
    bool m = (rank < k) || (id == CLS_ID_) || (id == EOS_ID_);
    mask_out[b * L_ + t] = m ? 1.0f : 0.0f;
    if (m) { int p = atomicAdd(&cnt, 1); slist[p] = t; }
  }
  __syncthreads();
  int c = cnt;
  for (int i = c + t; i < MAXSEL_TXT; i += blockDim.x) slist[i] = -1;
  __syncthreads();
  for (int i = t; i < MAXSEL_TXT; i += blockDim.x) sel[b * MAXSEL_TXT + i] = slist[i];
  if (t == 0) { cnt_out[b] = c; invc[b] = (c > 0) ? 1.0f / (float)c : 1.0f; }
}

// ---------------------------------------------------------------------------
// K4: masked mean pool over SELECTED rows only, via V_WMMA_F32_16X16X4_F32.
//   A (16x4 f32): weight 1/count replicated over M rows (0 for pad slots)
//   B (4x16 f32): gathered token rows, 16 contiguous columns per half-wave
// Each wave handles TWO 16-column tiles (32 columns) of one batch, reusing
// the A fragment + selection list across both WMMAs. D is a template constant
// so row offsets lower to shifts (no u64 mul in the hot loop); all offsets
// are 32-bit (max 197*1024 < 2^21).
// ---------------------------------------------------------------------------
template <int D_, int NTOK_, int MAXSEL_>
__global__ void k_pool_wmma(const float* __restrict__ tokens, const int* __restrict__ sel,
                            const int* __restrict__ cnt, const float* __restrict__ invc,
                            float* __restrict__ out) {
  int b  = blockIdx.y;
  int d0 = blockIdx.x * 32;     // two 16-wide tiles per wave
  int lane = threadIdx.x;       // 0..31, wave32
  int g = lane >> 4;            // half-wave group
  int n = lane & 15;            // output column within tile
  int dn = d0 + n;
  const float* tb = tokens + (size_t)b * (size_t)NTOK_ * (size_t)D_;
  const int2* slv = (const int2*)(sel + b * MAXSEL_);  // 8B aligned (k0 even)
  float w = invc[b];
  int c = cnt[b];
  int iters = (c + 3) >> 2;     // uniform per block; EXEC stays all-ones
  v8f acc0 = {}, acc1 = {};
  for (int it = 0; it < iters; ++it) {
    int2 ii = slv[it * 2 + g];  // sel[k0], sel[k0+1] with k0 = 4*it + 2*g
    v2f a;
    a.x = (ii.x >= 0) ? w : 0.0f;
    a.y = (ii.y >= 0) ? w : 0.0f;
    int r0 = (ii.x >= 0) ? ii.x : 0;   // pad slots: weight 0, safe row 0 address
    int r1 = (ii.y >= 0) ? ii.y : 0;
    int o0 = r0 * D_ + dn;             // D_ constexpr -> shifts
    int o1 = r1 * D_ + dn;
    v2f b0, b1;
    b0.x = tb[o0];
    b0.y = tb[o1];
    b1.x = tb[o0 + 16];
    b1.y = tb[o1 + 16];
    acc0 = __builtin_amdgcn_wmma_f32_16x16x4_f32(false, a, false, b0, (short)0, acc0,
                                                 false, false);
    acc1 = __builtin_amdgcn_wmma_f32_16x16x4_f32(false, a, false, b1, (short)0, acc1,
                                                 false, false);
  }
  if (lane < 16) {
    out[(size_t)b * D_ + dn]      = acc0[0];  // M=0 row of replicated result
    out[(size_t)b * D_ + dn + 16] = acc1[0];
  }
}

// ---------------------------------------------------------------------------
extern "C" void kernel_launch(void* const* d_in, const int* in_sizes, int n_in,
                              void* d_out, int out_size, void* d_ws, size_t ws_size,
                              hipStream_t stream) {
  const float* img_tokens  = (const float*)d_in[0];  // (B,N,Dv)
  const float* img_attn    = (const float*)d_in[1];  // (B,H,N,N)
  const float* text_tokens = (const float*)d_in[2];  // (B,L,Dt)
  const int*   text_ids    = (const int*)d_in[3];    // (B,L)

  float* out = (float*)d_out;
  float* out_img_mask = out;                               // B*N
  float* out_txt_mask = out + (size_t)B_ * N_;             // B*L
  float* out_img_feat = out_txt_mask + (size_t)B_ * L_;    // B*Dv
  float* out_txt_feat = out_img_feat + (size_t)B_ * DV_;   // B*Dt

  char* ws = (char*)d_ws;
  size_t o = 0;
  auto carve = [&](size_t bytes) -> char* {
    char* p = ws + o;
    o = (o + bytes + 255) & ~(size_t)255;
    return p;
  };
  float* pimp     = (float*)carve((size_t)B_ * N_ * 4);
  float* timp     = (float*)carve((size_t)B_ * L_ * 4);
  int*   img_sel  = (int*)carve((size_t)B_ * MAXSEL_IMG * 4);
  int*   img_cnt  = (int*)carve((size_t)B_ * 4);
  float* img_invc = (float*)carve((size_t)B_ * 4);
  int*   txt_sel  = (int*)carve((size_t)B_ * MAXSEL_TXT * 4);
  int*   txt_cnt  = (int*)carve((size_t)B_ * 4);
  float* txt_invc = (float*)carve((size_t)B_ * 4);
  (void)ws_size; (void)in_sizes; (void)n_in; (void)out_size;

  hipLaunchKernelGGL(k_patch_imp, dim3(B_), dim3(224), 0, stream, img_attn, pimp);
  hipLaunchKernelGGL(k_text_norm, dim3(L_, B_), dim3(192), 0, stream,
                     text_tokens, text_ids, timp);
  hipLaunchKernelGGL(k_img_select, dim3(B_), dim3(224), 0, stream,
                     pimp, out_img_mask, img_sel, img_cnt, img_invc);
  hipLaunchKernelGGL(k_txt_select, dim3(B_), dim3(128), 0, stream,
                     timp, text_ids, out_txt_mask, txt_sel, txt_cnt, txt_invc);
  hipLaunchKernelGGL((k_pool_wmma<DV_, N_, MAXSEL_IMG>), dim3(DV_ / 32, B_), dim3(32),
                     0, stream, img_tokens, img_sel, img_cnt, img_invc, out_img_feat);
  hipLaunchKernelGGL((k_pool_wmma<DT_, L_, MAXSEL_TXT>), dim3(DT_ / 32, B_), dim3(32),
                     0, stream, text_tokens, txt_sel, txt_cnt, txt_invc, out_txt_feat);
}